// SpatialNumDualDescriptorAB4_2740189135097
// MI455X (gfx1250) — compile-verified
//
#include <hip/hip_runtime.h>

// ---- problem constants (from reference) ----
#define MD        128            // M_DIM
#define LSZ       256            // L = 4^4
#define POS_PER_B 50625          // 15^4 windows per batch
#define TOTAL_ROWS 202500        // 4 * 15^4
#define NUM_TILES  12657         // ceil(TOTAL_ROWS / 16)

typedef __attribute__((ext_vector_type(2))) float v2f;
typedef __attribute__((ext_vector_type(4))) float v4f;
typedef __attribute__((ext_vector_type(8))) float v8f;

// D = A(16x4 pos x i) * B(4x16 i x j) + C, accumulated over K=128 in steps of 4.
// Per workgroup: 16 output rows x 128 output cols; 8 waves, one 16-col slice each.
__global__ void __launch_bounds__(256)
sndd_ab4_kernel(const float* __restrict__ vec,     // (4,16,16,16,16,128)
                const float* __restrict__ Mmat,    // (128,128)
                const float* __restrict__ Acoeff,  // (128,256)
                const float* __restrict__ Bbasis,  // (256,128)
                float* __restrict__ out)           // (4,15^4,128)
{
    __shared__ float sP[16][132];   // pooled A-panel; 132 floats = 528 B row pitch
                                    // (528 % 256 != 0 -> conflict-free, 16B-aligned rows)
    __shared__ int   sBase[16];     // window-origin element offset into vec
    __shared__ int   sIdx[16];      // basis index idx(pos)
    __shared__ int   sValid[16];    // row < TOTAL_ROWS

    const int tid  = threadIdx.x;
    const int wave = tid >> 5;      // 0..7 : owns output cols [16*wave, 16*wave+16)
    const int lane = tid & 31;
    const int nlo  = lane & 15;     // N (and A-row) index
    const int hi   = lane >> 4;     // K sub-select per ISA layout
    const int jcol = wave * 16 + nlo;

    // Preload this wave's B panel: B(K=i, N=jcol) = M[jcol, i]; K = 4*kb + 2*hi + v
    float breg[64];
#pragma unroll
    for (int kb = 0; kb < 32; ++kb) {
        breg[2 * kb + 0] = Mmat[jcol * MD + 4 * kb + 2 * hi + 0];
        breg[2 * kb + 1] = Mmat[jcol * MD + 4 * kb + 2 * hi + 1];
    }

    for (int tile = blockIdx.x; tile < NUM_TILES; tile += gridDim.x) {
        const int r0 = tile * 16;

        // ---- pre-phase: per-row window origin + basis index ----
        if (tid < 16) {
            int r     = r0 + tid;
            int valid = (r < TOTAL_ROWS);
            int rr    = valid ? r : (TOTAL_ROWS - 1);
            int b     = rr / POS_PER_B;
            int pos   = rr - b * POS_PER_B;
            int k1 = pos / 3375;  int t = pos - k1 * 3375;
            int k2 = t / 225;     t -= k2 * 225;
            int k3 = t / 15;      int k4 = t - k3 * 15;
            sBase[tid]  = ((((b * 16 + k1) * 16 + k2) * 16 + k3) * 16 + k4) * MD;
            sIdx[tid]   = (((k1 & 3) * 4 + (k2 & 3)) * 4 + (k3 & 3)) * 4 + (k4 & 3);
            sValid[tid] = valid;
        }
        __syncthreads();

        // ---- pooling: 16 rows x 128 channels, 2^4 window mean, float4 lanes ----
        // 512 groups of 4 channels; each thread owns 2 groups -> 32x global_load_b128
#pragma unroll
        for (int it = 0; it < 2; ++it) {
            int e = tid + it * 256;          // 0..511
            int p = e >> 5;                  // row in tile
            int g = e & 31;                  // 4-channel group
            const v4f* src = (const v4f*)(vec + sBase[p] + 4 * g);
            v4f s = {};
#pragma unroll
            for (int d1 = 0; d1 < 2; ++d1)
#pragma unroll
                for (int d2 = 0; d2 < 2; ++d2)
#pragma unroll
                    for (int d3 = 0; d3 < 2; ++d3)
#pragma unroll
                        for (int d4 = 0; d4 < 2; ++d4)
                            s += src[d1 * 131072 + d2 * 8192 + d3 * 512 + d4 * 32];
            s *= 0.0625f;
            *(v4f*)(&sP[p][4 * g]) = s;      // aligned ds_store_b128
        }
        __syncthreads();

        // ---- WMMA: C(16x16) += A(16x4) * B(4x16), 32 steps over K=128 ----
        v8f acc = {};
#pragma unroll
        for (int kb = 0; kb < 32; ++kb) {
            v2f a;
            a.x = sP[nlo][4 * kb + 2 * hi + 0];   // A row = nlo, K = 4*kb + 2*hi + v
            a.y = sP[nlo][4 * kb + 2 * hi + 1];
            v2f b;
            b.x = breg[2 * kb + 0];
            b.y = breg[2 * kb + 1];
            acc = __builtin_amdgcn_wmma_f32_16x16x4_f32(
                false, a, false, b, (short)0, acc, false, false);
        }

        // ---- epilogue: out = Nk - agg ; C/D row = v + 8*hi, col = jcol ----
        // Non-temporal stores: output is write-once; keep vec L2-resident.
#pragma unroll
        for (int v = 0; v < 8; ++v) {
            int p = v + 8 * hi;
            if (sValid[p]) {
                int idxp = sIdx[p];
                float nk = Acoeff[jcol * LSZ + idxp] * Bbasis[idxp * MD + jcol];
                __builtin_nontemporal_store(nk - acc[v],
                                            &out[(size_t)(r0 + p) * MD + jcol]);
            }
        }
        __syncthreads();   // protect sP/sBase before next tile overwrites
    }
}

extern "C" void kernel_launch(void* const* d_in, const int* in_sizes, int n_in,
                              void* d_out, int out_size, void* d_ws, size_t ws_size,
                              hipStream_t stream) {
    (void)in_sizes; (void)n_in; (void)out_size; (void)d_ws; (void)ws_size;
    const float* vec    = (const float*)d_in[0];
    const float* Mmat   = (const float*)d_in[1];
    const float* Acoeff = (const float*)d_in[2];
    const float* Bbasis = (const float*)d_in[3];
    float* out = (float*)d_out;

    dim3 block(256);
    dim3 grid(NUM_TILES);
    hipLaunchKernelGGL(sndd_ab4_kernel, grid, block, 0, stream,
                       vec, Mmat, Acoeff, Bbasis, out);
}